// WeightedPreprocessing_73426760892671
// MI455X (gfx1250) — compile-verified
//
#include <hip/hip_runtime.h>
#include <hip/hip_bf16.h>
#include <stdint.h>

#define DEG 64
#define THREADS 256
#define NODES_PER_BLOCK 256
#define ROW_BYTES (DEG * 4)                         // 256 B per node row
#define TILE_BYTES (NODES_PER_BLOCK * ROW_BYTES)    // 64 KB LDS tile
#define CHUNK_BYTES (THREADS * 16)                  // 4 KB per async chunk
#define CHUNKS (TILE_BYTES / CHUNK_BYTES)           // 16

#if __has_builtin(__builtin_amdgcn_global_load_async_to_lds_b128) && \
    __has_builtin(__builtin_amdgcn_global_store_async_from_lds_b128) && \
    __has_builtin(__builtin_amdgcn_s_wait_asynccnt)
#define HAVE_ASYNC_LDS 1
#else
#define HAVE_ASYNC_LDS 0
#endif

// Builtin parameter type per hipcc diagnostic:
//   "__attribute__((__vector_size__(4 * sizeof(int)))) int *"
typedef int v4i __attribute__((vector_size(16)));

// Poisson-binomial DP chunk: 16 edge steps with a fixed update ceiling BOUND.
// Updating entries above the live prefix is a provable no-op (they stay 0),
// so a constant bound per 16-step chunk is exact while keeping all dp[]
// indices compile-time constant -> dp lives entirely in VGPRs.
template <int BOUND>
__device__ __forceinline__ void dp_chunk(float* dp, const float* wrow) {
#pragma unroll
  for (int kk = 0; kk < 16; ++kk) {
    const float p = wrow[kk];        // ds_load from this lane's LDS row
    const float q = 1.0f - p;
#pragma unroll
    for (int d = BOUND; d >= 1; --d)       // descending: uses pre-update dp[d-1]
      dp[d] = __builtin_fmaf(dp[d], q, dp[d - 1] * p);
    dp[0] *= q;
  }
}

__global__ __launch_bounds__(THREADS) void poisson_binom_dp_kernel(
    const float* __restrict__ w, float* __restrict__ out, int n_nodes) {
  __shared__ float smem[TILE_BYTES / 4];

  const int tid = threadIdx.x;
  const int node0 = blockIdx.x * NODES_PER_BLOCK;
  const int nvalid = min(NODES_PER_BLOCK, n_nodes - node0);
  if (nvalid <= 0) return;
  const int valid_bytes = nvalid * ROW_BYTES;

  // First tuple output: degree indices 1..64 (stored as float in d_out[0:64]).
  if (blockIdx.x == 0 && tid < DEG) out[tid] = (float)(tid + 1);

  // ---- Stage-in: coalesced async global -> LDS (16 B per lane per chunk) ----
  const char* gsrc = (const char*)(w + (size_t)node0 * DEG);
#if HAVE_ASYNC_LDS
#pragma unroll
  for (int i = 0; i < CHUNKS; ++i) {
    const int off = i * CHUNK_BYTES + tid * 16;
    if (off < valid_bytes) {
      __builtin_amdgcn_global_load_async_to_lds_b128(
          (v4i*)(gsrc + off), (v4i*)((char*)smem + off), 0, 0);
    }
  }
  __builtin_amdgcn_s_wait_asynccnt(0);
#else
#pragma unroll
  for (int i = 0; i < CHUNKS; ++i) {
    const int off = i * CHUNK_BYTES + tid * 16;
    if (off < valid_bytes)
      *(float4*)((char*)smem + off) = *(const float4*)(gsrc + off);
  }
#endif
  __syncthreads();

  // ---- Per-lane DP over this lane's 64 weights (register-resident state) ----
  float dp[DEG + 1];
#pragma unroll
  for (int d = 0; d <= DEG; ++d) dp[d] = 0.0f;
  dp[0] = 1.0f;

  if (tid < nvalid) {
    const float* wrow = &smem[tid * DEG];
    dp_chunk<16>(dp, wrow + 0);
    dp_chunk<32>(dp, wrow + 16);
    dp_chunk<48>(dp, wrow + 32);
    dp_chunk<64>(dp, wrow + 48);
    // Reuse own LDS row as the staging buffer for the output distribution.
    float* orow = &smem[tid * DEG];
#pragma unroll
    for (int d = 1; d <= DEG; ++d) orow[d - 1] = dp[d];
  }
  __syncthreads();

  // ---- Stage-out: coalesced async LDS -> global ----
  char* gdst = (char*)(out + DEG + (size_t)node0 * DEG);  // +64 floats = 256 B, 16B-aligned
#if HAVE_ASYNC_LDS
#pragma unroll
  for (int i = 0; i < CHUNKS; ++i) {
    const int off = i * CHUNK_BYTES + tid * 16;
    if (off < valid_bytes) {
      __builtin_amdgcn_global_store_async_from_lds_b128(
          (v4i*)(gdst + off), (v4i*)((char*)smem + off), 0, 0);
    }
  }
  __builtin_amdgcn_s_wait_asynccnt(0);
#else
#pragma unroll
  for (int i = 0; i < CHUNKS; ++i) {
    const int off = i * CHUNK_BYTES + tid * 16;
    if (off < valid_bytes)
      *(float4*)(gdst + off) = *(const float4*)((char*)smem + off);
  }
#endif
}

extern "C" void kernel_launch(void* const* d_in, const int* in_sizes, int n_in,
                              void* d_out, int out_size, void* d_ws, size_t ws_size,
                              hipStream_t stream) {
  // Inputs (setup_inputs order): edge_weights f32[N*64], edge_index i32[2*N*64]
  // (unused: src is implicit-contiguous, dst irrelevant), num_nodes, max_degree.
  const float* w = (const float*)d_in[0];
  const int n_nodes = in_sizes[0] / DEG;
  float* out = (float*)d_out;

  const int blocks = (n_nodes + NODES_PER_BLOCK - 1) / NODES_PER_BLOCK;
  poisson_binom_dp_kernel<<<blocks, THREADS, 0, stream>>>(w, out, n_nodes);
}